// EnzymeGNN_62938450755744
// MI455X (gfx1250) — compile-verified
//
#include <hip/hip_runtime.h>
#include <math.h>

typedef __attribute__((ext_vector_type(2))) float v2f;
typedef __attribute__((ext_vector_type(8))) float v8f;

#define EPSV 1e-5f
#define KCMAX 128         // K-chunk staged in LDS
#define BPAD  4           // LDS row pad (stride = KC+4 -> bank-conflict-free)

__device__ __forceinline__ float gelu_exact(float x) {
    return 0.5f * x * (1.0f + erff(x * 0.7071067811865475f));
}

// ---------------- degree / norm ----------------
__global__ void k_init_deg(float* deg, int n) {
    int i = blockIdx.x * blockDim.x + threadIdx.x;
    if (i < n) deg[i] = 1.0f;                 // self-loop contributes 1
}
__global__ void k_edge_deg(const long long* __restrict__ ei, float* deg, int E) {
    int e = blockIdx.x * blockDim.x + threadIdx.x;
    if (e < E) atomicAdd(&deg[(int)ei[E + e]], 1.0f);
}
__global__ void k_rsqrt(float* deg, int n) {
    int i = blockIdx.x * blockDim.x + threadIdx.x;
    if (i < n) deg[i] = rsqrtf(deg[i]);
}

// ---- WMMA fp32 GEMM: C[M,N] = A[M,K] @ B[K,N] (+bias)(+gelu) ----
// block = 256 thr = 8 waves; each wave owns one 16-row m-tile and
// accumulates FOUR 16x16 n-subtiles (64 cols) per A-fragment load.
// K is compile-time -> static trip counts, unrolled & pipelined loops.
template <int K>
__global__ __launch_bounds__(256) void k_gemm_wmma(
    const float* __restrict__ A, const float* __restrict__ B,
    float* __restrict__ C, const float* __restrict__ bias,
    int M, int N, int act)
{
    constexpr int KCC = (K < KCMAX) ? K : KCMAX;   // chunk size (64 or 128)
    constexpr int SB  = KCC + BPAD;                // LDS stride
    __shared__ float Bs[64 * SB];

    const int colbase = blockIdx.y * 64;
    const int wave  = threadIdx.x >> 5;
    const int lane  = threadIdx.x & 31;
    const int mtile = blockIdx.x * 8 + wave;
    const bool active = (mtile * 16) < M;          // wave-uniform

    const int half = lane >> 4;                    // 0: lanes 0-15, 1: 16-31
    const int koff = half << 1;                    // K offset 0 or 2
    const int n    = lane & 15;

    const int arow = active ? (mtile * 16 + n) : 0;
    const float* Arow = A + (size_t)arow * K + koff;
    const float* Bp   = Bs + n * SB + koff;

    v8f acc0 = {}, acc1 = {}, acc2 = {}, acc3 = {};

    #pragma unroll 1
    for (int k0 = 0; k0 < K; k0 += KCC) {
        // cooperative load of B chunk: B[k0+kk][colbase+c] -> Bs[c][kk]
        #pragma unroll
        for (int idx = 0; idx < (KCC << 6); idx += 256) {
            int t  = idx + threadIdx.x;
            int c  = t & 63;
            int kk = t >> 6;
            int col = colbase + c;
            float v = (col < N) ? B[(size_t)(k0 + kk) * N + col] : 0.0f;
            Bs[c * SB + kk] = v;
        }
        __syncthreads();

        #pragma unroll 4
        for (int k = 0; k < KCC; k += 4) {
            v2f a  = *(const v2f*)(Arow + k0 + k);        // global_load_b64
            v2f b0 = *(const v2f*)(Bp + 0 * 16 * SB + k); // ds_load_b64
            v2f b1 = *(const v2f*)(Bp + 1 * 16 * SB + k);
            v2f b2 = *(const v2f*)(Bp + 2 * 16 * SB + k);
            v2f b3 = *(const v2f*)(Bp + 3 * 16 * SB + k);
            acc0 = __builtin_amdgcn_wmma_f32_16x16x4_f32(false, a, false, b0, (short)0, acc0, false, false);
            acc1 = __builtin_amdgcn_wmma_f32_16x16x4_f32(false, a, false, b1, (short)0, acc1, false, false);
            acc2 = __builtin_amdgcn_wmma_f32_16x16x4_f32(false, a, false, b2, (short)0, acc2, false, false);
            acc3 = __builtin_amdgcn_wmma_f32_16x16x4_f32(false, a, false, b3, (short)0, acc3, false, false);
        }
        __syncthreads();
    }

    if (!active) return;
    const int rbase = mtile * 16 + half * 8;
    v8f accs[4] = {acc0, acc1, acc2, acc3};
    #pragma unroll
    for (int j = 0; j < 4; ++j) {
        int col = colbase + j * 16 + n;
        if (col < N) {
            float bi = bias ? bias[col] : 0.0f;
            #pragma unroll
            for (int r = 0; r < 8; ++r) {
                float v = accs[j][r] + bi;
                if (act == 1) v = gelu_exact(v);
                C[(size_t)(rbase + r) * N + col] = v;
            }
        }
    }
}

// ---------------- GCN aggregation ----------------
// self-loop term initializes accumulator: H[i] = T[i] * dis[i]^2
__global__ void k_self_agg(const float* __restrict__ T, float* __restrict__ H,
                           const float* __restrict__ dis, int N, int F, int lgt) {
    int t = blockIdx.x * blockDim.x + threadIdx.x;
    int tpe = F >> 2;
    if (t >= N * tpe) return;
    int i = t >> lgt;
    int c = t & (tpe - 1);
    float s = dis[i]; s *= s;
    float4 v = *(const float4*)(T + (size_t)i * F + (c << 2));
    v.x *= s; v.y *= s; v.z *= s; v.w *= s;
    *(float4*)(H + (size_t)i * F + (c << 2)) = v;
}

__global__ void k_edge_agg(const float* __restrict__ T, float* __restrict__ H,
                           const long long* __restrict__ ei, const float* __restrict__ dis,
                           int E, int F, int lgt) {
    long long t = (long long)blockIdx.x * blockDim.x + threadIdx.x;
    int tpe = F >> 2;
    if (t >= (long long)E * tpe) return;
    int e = (int)(t >> lgt);
    int c = (int)(t & (tpe - 1));
    int s = (int)ei[e];
    int d = (int)ei[E + e];
    float nm = dis[s] * dis[d];
    float4 v = *(const float4*)(T + (size_t)s * F + (c << 2));
    float* out = H + (size_t)d * F + (c << 2);
    atomicAdd(out + 0, v.x * nm);
    atomicAdd(out + 1, v.y * nm);
    atomicAdd(out + 2, v.z * nm);
    atomicAdd(out + 3, v.w * nm);
}

// bias + batchnorm + relu (F is a power of two)
__global__ void k_bias_bn_relu(float* __restrict__ H, const float* __restrict__ b,
                               const float* __restrict__ g, const float* __restrict__ be,
                               const float* __restrict__ m, const float* __restrict__ v,
                               int N, int F) {
    long long t = (long long)blockIdx.x * blockDim.x + threadIdx.x;
    if (t >= (long long)N * F) return;
    int f = (int)(t & (long long)(F - 1));
    float x = H[t];
    x = (x + b[f] - m[f]) * rsqrtf(v[f] + EPSV) * g[f] + be[f];
    H[t] = fmaxf(x, 0.0f);
}

// ---------------- pooling ----------------
__global__ void k_pool_init(float* pmean, unsigned* pmax, float* cnts, int GF, int G) {
    int i = blockIdx.x * blockDim.x + threadIdx.x;
    if (i < GF) { pmean[i] = 0.0f; pmax[i] = 0u; }
    if (i < G) cnts[i] = 0.0f;
}
__global__ void k_count(const long long* __restrict__ batch, float* cnts, int N) {
    int i = blockIdx.x * blockDim.x + threadIdx.x;
    if (i < N) atomicAdd(&cnts[(int)batch[i]], 1.0f);
}
__global__ void k_pool_accum(const float* __restrict__ H, const long long* __restrict__ batch,
                             float* pmean, unsigned* pmax, int N) {
    int t = blockIdx.x * blockDim.x + threadIdx.x;   // F=256, tpe=64
    if (t >= N * 64) return;
    int i = t >> 6;
    int c = t & 63;
    int g = (int)batch[i];
    float4 v = *(const float4*)(H + (size_t)i * 256 + (c << 2));
    float*    pm = pmean + (size_t)g * 256 + (c << 2);
    unsigned* px = pmax  + (size_t)g * 256 + (c << 2);
    atomicAdd(pm + 0, v.x); atomicAdd(pm + 1, v.y);
    atomicAdd(pm + 2, v.z); atomicAdd(pm + 3, v.w);
    atomicMax(px + 0, __float_as_uint(v.x));          // values >= 0 post-ReLU:
    atomicMax(px + 1, __float_as_uint(v.y));          // uint order == float order
    atomicMax(px + 2, __float_as_uint(v.z));
    atomicMax(px + 3, __float_as_uint(v.w));
}
__global__ void k_pool_final(const float* __restrict__ pmean, const unsigned* __restrict__ pmax,
                             const float* __restrict__ cnts, float* __restrict__ z, int G) {
    int t = blockIdx.x * blockDim.x + threadIdx.x;
    if (t >= G * 512) return;
    int g = t >> 9;
    int f = t & 511;
    float out;
    if (f < 256) out = pmean[g * 256 + f] / fmaxf(cnts[g], 1.0f);
    else         out = __uint_as_float(pmax[g * 256 + (f - 256)]);
    z[t] = out;
}

// ---------------- host side ----------------
extern "C" void kernel_launch(void* const* d_in, const int* in_sizes, int n_in,
                              void* d_out, int out_size, void* d_ws, size_t ws_size,
                              hipStream_t stream) {
    const float*     x     = (const float*)d_in[0];
    const long long* ei    = (const long long*)d_in[1];
    const long long* batch = (const long long*)d_in[2];
    const float *W1=(const float*)d_in[3],  *b1=(const float*)d_in[4],  *g1=(const float*)d_in[5],
                *be1=(const float*)d_in[6], *m1=(const float*)d_in[7],  *v1=(const float*)d_in[8];
    const float *W2=(const float*)d_in[9],  *b2=(const float*)d_in[10], *g2=(const float*)d_in[11],
                *be2=(const float*)d_in[12],*m2=(const float*)d_in[13], *v2=(const float*)d_in[14];
    const float *W3=(const float*)d_in[15], *b3=(const float*)d_in[16], *g3=(const float*)d_in[17],
                *be3=(const float*)d_in[18],*m3=(const float*)d_in[19], *v3=(const float*)d_in[20];
    const float *Wm1=(const float*)d_in[21], *bm1=(const float*)d_in[22];
    const float *Wm2=(const float*)d_in[23], *bm2=(const float*)d_in[24];
    const float *Wm3=(const float*)d_in[25], *bm3=(const float*)d_in[26];

    const int N = in_sizes[0] / 64;      // 50000 (multiple of 16)
    const int E = in_sizes[1] / 2;       // 800000
    const int G = 512;

    // workspace layout (floats)
    float* ws = (float*)d_ws;
    size_t off = 0;
    float* dis  = ws + off;  off += 50176;                 // N padded
    float* bufT = ws + off;  off += (size_t)N * 256;
    float* bufH = ws + off;  off += (size_t)N * 256;
    float* pmean = ws + off; off += (size_t)G * 256;
    unsigned* pmax = (unsigned*)(ws + off); off += (size_t)G * 256;
    float* cnts = ws + off;  off += 1024;
    float* z    = ws + off;  off += (size_t)G * 512;
    float* z1   = ws + off;  off += (size_t)G * 256;
    float* z2   = ws + off;  off += (size_t)G * 128;

    const int B = 256;
    auto cdiv = [](long long a, long long b) { return (int)((a + b - 1) / b); };

    // degree -> dis = rsqrt(deg)
    k_init_deg<<<cdiv(N, B), B, 0, stream>>>(dis, N);
    k_edge_deg<<<cdiv(E, B), B, 0, stream>>>(ei, dis, E);
    k_rsqrt  <<<cdiv(N, B), B, 0, stream>>>(dis, N);

    const int mtiles = N / 16;                      // 3125
    // ---- GCN layer 1: K=64 -> 128 ----
    {
        const int F = 128, lgt = 5;
        dim3 grid(cdiv(mtiles, 8), cdiv(F, 64));
        k_gemm_wmma<64><<<grid, B, 0, stream>>>(x, W1, bufT, nullptr, N, F, 0);
        k_self_agg<<<cdiv((long long)N * (F / 4), B), B, 0, stream>>>(bufT, bufH, dis, N, F, lgt);
        k_edge_agg<<<cdiv((long long)E * (F / 4), B), B, 0, stream>>>(bufT, bufH, ei, dis, E, F, lgt);
        k_bias_bn_relu<<<cdiv((long long)N * F, B), B, 0, stream>>>(bufH, b1, g1, be1, m1, v1, N, F);
    }
    // ---- GCN layer 2: K=128 -> 256 ----
    {
        const int F = 256, lgt = 6;
        dim3 grid(cdiv(mtiles, 8), cdiv(F, 64));
        k_gemm_wmma<128><<<grid, B, 0, stream>>>(bufH, W2, bufT, nullptr, N, F, 0);
        k_self_agg<<<cdiv((long long)N * (F / 4), B), B, 0, stream>>>(bufT, bufH, dis, N, F, lgt);
        k_edge_agg<<<cdiv((long long)E * (F / 4), B), B, 0, stream>>>(bufT, bufH, ei, dis, E, F, lgt);
        k_bias_bn_relu<<<cdiv((long long)N * F, B), B, 0, stream>>>(bufH, b2, g2, be2, m2, v2, N, F);
    }
    // ---- GCN layer 3: K=256 -> 256 ----
    {
        const int F = 256, lgt = 6;
        dim3 grid(cdiv(mtiles, 8), cdiv(F, 64));
        k_gemm_wmma<256><<<grid, B, 0, stream>>>(bufH, W3, bufT, nullptr, N, F, 0);
        k_self_agg<<<cdiv((long long)N * (F / 4), B), B, 0, stream>>>(bufT, bufH, dis, N, F, lgt);
        k_edge_agg<<<cdiv((long long)E * (F / 4), B), B, 0, stream>>>(bufT, bufH, ei, dis, E, F, lgt);
        k_bias_bn_relu<<<cdiv((long long)N * F, B), B, 0, stream>>>(bufH, b3, g3, be3, m3, v3, N, F);
    }

    // ---- pooling: mean + max per graph -> z[G,512] ----
    k_pool_init <<<cdiv(G * 256, B), B, 0, stream>>>(pmean, pmax, cnts, G * 256, G);
    k_count     <<<cdiv(N, B), B, 0, stream>>>(batch, cnts, N);
    k_pool_accum<<<cdiv((long long)N * 64, B), B, 0, stream>>>(bufH, batch, pmean, pmax, N);
    k_pool_final<<<cdiv(G * 512, B), B, 0, stream>>>(pmean, pmax, cnts, z, G);

    // ---- MLP head (bias + exact GELU fused) ----
    {
        dim3 grid(cdiv(G / 16, 8), cdiv(256, 64));  // 512x512 @ 512x256
        k_gemm_wmma<512><<<grid, B, 0, stream>>>(z, Wm1, z1, bm1, G, 256, 1);
    }
    {
        dim3 grid(cdiv(G / 16, 8), cdiv(128, 64));  // 512x256 @ 256x128
        k_gemm_wmma<256><<<grid, B, 0, stream>>>(z1, Wm2, z2, bm2, G, 128, 1);
    }
    {
        dim3 grid(cdiv(G / 16, 8), 1);              // 512x128 @ 128x6 (masked cols)
        k_gemm_wmma<128><<<grid, B, 0, stream>>>(z2, Wm3, (float*)d_out, bm3, G, 6, 0);
    }
}